// MVFeedforwardBlock_30313879175206
// MI455X (gfx1250) — compile-verified
//
#include <hip/hip_runtime.h>
#include <math.h>

// ---------------------------------------------------------------------------
// MI455X / gfx1250 fused PGA Cl(3,0,1) MV-feedforward block.
//   proj  = W1 * x + b1                 (WMMA f32 16x16x4, K=16 -> 4 chained)
//   g     = gelu(proj0) * proj          (VALU + 1 shuffle broadcast)
//   gp    = Cayley(x, g)                (per-lane sparse, 192 +-1 FMAs)
//   join  = x15 * DualWedge(x, g)       (per-lane sparse,  81 +-1 FMAs)
//   out   = W2g * gp + W2j * join + b   (8 accumulating WMMAs)
// One wave owns a tile of 16 multivectors; LDS slices are wave-private so no
// barriers are needed (same-wave DS ops are in-order on CDNA5).
// ---------------------------------------------------------------------------

typedef __attribute__((ext_vector_type(2))) float v2f;
typedef __attribute__((ext_vector_type(8))) float v8f;

// ------------------------- compile-time Cayley tables ----------------------
namespace pga {
// blade order: 1,e0,e1,e2,e3,e01,e02,e03,e12,e13,e23,e012,e013,e023,e123,e0123
constexpr unsigned MASK[16] = {0u,1u,2u,4u,8u,3u,5u,9u,6u,10u,12u,7u,11u,13u,14u,15u};
constexpr int      IDX[16]  = {0,1,2,5,3,6,8,11,4,7,9,12,10,13,14,15}; // mask -> index

constexpr int rsign(unsigned a, unsigned b) {  // reorder sign of blade product a*b
  int s = 0;
  for (int k = 0; k < 4; ++k)
    if ((b >> k) & 1u)
      for (int j = k + 1; j < 4; ++j)
        if ((a >> j) & 1u) ++s;
  return (s & 1) ? -1 : 1;
}
struct Tab { signed char k[16][16]; signed char s[16][16]; };
constexpr Tab make_gp() {   // geometric product, metric e0^2=0, e1..e3^2=+1
  Tab t{};
  for (int i = 0; i < 16; ++i)
    for (int j = 0; j < 16; ++j) {
      unsigned a = MASK[i], b = MASK[j];
      if (a & b & 1u) { t.k[i][j] = 0; t.s[i][j] = 0; }     // e0*e0 = 0
      else { t.k[i][j] = (signed char)IDX[a ^ b];
             t.s[i][j] = (signed char)rsign(a, b); }
    }
  return t;
}
constexpr Tab make_join() { // undual(dual(x) ^ dual(y)), right complement
  Tab t{};
  for (int i = 0; i < 16; ++i)
    for (int j = 0; j < 16; ++j) {
      unsigned a = MASK[i], b = MASK[j];
      unsigned ca = a ^ 15u, cb = b ^ 15u;
      if (ca & cb) { t.k[i][j] = 0; t.s[i][j] = 0; continue; }
      int sa = rsign(a, ca), sb = rsign(b, cb), sw = rsign(ca, cb);
      unsigned m = ca ^ cb, km = m ^ 15u;
      int sk = rsign(km, m);                     // undual sign (D is +-1 perm)
      t.k[i][j] = (signed char)IDX[km];
      t.s[i][j] = (signed char)(sa * sb * sw * sk);
    }
  return t;
}
constexpr Tab GP = make_gp();
constexpr Tab JN = make_join();
} // namespace pga

// index -> blade bitmask, packed 4 bits per index
#define MASKPACK 0xFEDB7CA695384210ULL

// W[m][K] of an MVLinear: diag = w[grade(m)]; e0-branch: if blade_K has no e0
// and blade_m == blade_K|e0 (and grade(m)<=3) add v[grade(m)].
__device__ __forceinline__ float wentry(int m, int K,
                                        float w0, float w1, float w2, float w3, float w4,
                                        float vg1, float vg2, float vg3) {
  unsigned mm = (unsigned)(MASKPACK >> (4 * m)) & 15u;
  unsigned mk = (unsigned)(MASKPACK >> (4 * K)) & 15u;
  int gm = __popc(mm);
  float wg = (gm == 0) ? w0 : (gm == 1) ? w1 : (gm == 2) ? w2 : (gm == 3) ? w3 : w4;
  float r = (m == K) ? wg : 0.0f;
  if (((mk & 1u) == 0u) && (mm == (mk | 1u)) && (mm != 15u)) {
    float vg = (gm == 1) ? vg1 : (gm == 2) ? vg2 : vg3;
    r += vg;
  }
  return r;
}

// D(16x16) = A(16xK=16) * B(K=16 x 16) + C via 4 chained V_WMMA_F32_16X16X4_F32.
// A-frag (per doc): lane l holds M=l%16, K = 4s + vgpr + 2*(l/16).
// B-frag (B row striped across lanes): lane l holds N=l%16, K = 4s + vgpr + 2*(l/16).
// B is read element-major from LDS (elem stride LS), so b = float2 @ [e*LS + Kbase].
#define LS 20   // LDS row stride in floats: 16B-aligned float4 rows, conflict-free cols

__device__ __forceinline__ v8f gemm_acc(const float* Bbase, int e, int h,
                                        const v2f* a, v8f c) {
#pragma unroll
  for (int s = 0; s < 4; ++s) {
    v2f b = *(const v2f*)(Bbase + e * LS + 4 * s + 2 * h);
    c = __builtin_amdgcn_wmma_f32_16x16x4_f32(false, a[s], false, b,
                                              (short)0, c, false, false);
  }
  return c;
}

__global__ __launch_bounds__(256) void mv_ffn_kernel(
    const float* __restrict__ x,
    const float* __restrict__ w1p,  const float* __restrict__ v1p,  const float* __restrict__ b1p,
    const float* __restrict__ w2gp, const float* __restrict__ v2gp, const float* __restrict__ b2gp,
    const float* __restrict__ w2jp, const float* __restrict__ v2jp, const float* __restrict__ b2jp,
    float* __restrict__ out, int tiles) {
  __shared__ __align__(16) float smem[8 * 4 * 16 * LS];  // 8 waves x (X,G,GP,J) tiles

  const int wave = threadIdx.x >> 5;
  const int lane = threadIdx.x & 31;
  const int ee   = lane & 15;   // element within tile (A: row M; B/C: col N)
  const int h    = lane >> 4;   // half-wave (selects K/M offset)

  float* XB  = &smem[wave * (4 * 16 * LS)];
  float* GB  = XB + 16 * LS;
  float* GPB = XB + 32 * LS;
  float* JB  = XB + 48 * LS;

  // uniform scalar weight loads
  const float w10 = w1p[0],  w11 = w1p[1],  w12 = w1p[2],  w13 = w1p[3],  w14 = w1p[4];
  const float v11 = v1p[1],  v12 = v1p[2],  v13 = v1p[3];
  const float g10 = w2gp[0], g11 = w2gp[1], g12 = w2gp[2], g13 = w2gp[3], g14 = w2gp[4];
  const float h11 = v2gp[1], h12 = v2gp[2], h13 = v2gp[3];
  const float j10 = w2jp[0], j11 = w2jp[1], j12 = w2jp[2], j13 = w2jp[3], j14 = w2jp[4];
  const float k11 = v2jp[1], k12 = v2jp[2], k13 = v2jp[3];
  const float b1v   = b1p[0];
  const float bias2 = b2gp[0] + b2jp[0];

  // Hoist weight A-fragments out of the tile loop (loop-invariant, 24 VGPRs).
  v2f a1[4], a2g[4], a2j[4];
#pragma unroll
  for (int s = 0; s < 4; ++s) {
    const int K = 4 * s + 2 * h;
    a1[s]  = (v2f){ wentry(ee, K,     w10,w11,w12,w13,w14, v11,v12,v13),
                    wentry(ee, K + 1, w10,w11,w12,w13,w14, v11,v12,v13) };
    a2g[s] = (v2f){ wentry(ee, K,     g10,g11,g12,g13,g14, h11,h12,h13),
                    wentry(ee, K + 1, g10,g11,g12,g13,g14, h11,h12,h13) };
    a2j[s] = (v2f){ wentry(ee, K,     j10,j11,j12,j13,j14, k11,k12,k13),
                    wentry(ee, K + 1, j10,j11,j12,j13,j14, k11,k12,k13) };
  }

  const int le = lane >> 2, lc = (lane & 3) * 4;  // coalesced <-> padded LDS map

  for (long long tile = (long long)blockIdx.x * 8 + wave; tile < tiles;
       tile += (long long)gridDim.x * 8) {
    // ---- coalesced load of 16 multivectors (1 KiB) into padded LDS ----
    const float4* xin = (const float4*)(x + tile * 256);
    float4 p0 = xin[lane];
    float4 p1 = xin[lane + 32];
    *(float4*)(XB + le * LS + lc)       = p0;
    *(float4*)(XB + (le + 8) * LS + lc) = p1;
    __asm__ volatile("" ::: "memory");

    // ---- proj = W1 * X^T  (component-major fragments: M=comp, N=elem) ----
    v8f P = {0.f,0.f,0.f,0.f,0.f,0.f,0.f,0.f};
    P = gemm_acc(XB, ee, h, a1, P);
    if (h == 0) P[0] += b1v;                       // bias on scalar component

    // ---- gate by exact gelu of scalar component, broadcast across M ----
    float s0 = __shfl(P[0], ee, 32);               // P(M=0, N=ee) lives in lane ee
    float gg = 0.5f * s0 * (1.0f + erff(s0 * 0.70710678118654752f));
    // store gated G element-major (comp = r + 8h)
#pragma unroll
    for (int r = 0; r < 8; r += 2) {
      v2f gp2 = { gg * P[r], gg * P[r + 1] };
      *(v2f*)(GB + ee * LS + 8 * h + r) = gp2;
    }
    __asm__ volatile("" ::: "memory");

    // ---- per-lane sparse bilinears: lanes 0-15 gp, lanes 16-31 join ----
    float xr[16], gr[16];
#pragma unroll
    for (int k = 0; k < 16; k += 4) {
      float4 t = *(const float4*)(XB + ee * LS + k);
      xr[k] = t.x; xr[k + 1] = t.y; xr[k + 2] = t.z; xr[k + 3] = t.w;
      float4 u = *(const float4*)(GB + ee * LS + k);
      gr[k] = u.x; gr[k + 1] = u.y; gr[k + 2] = u.z; gr[k + 3] = u.w;
    }
    float acc[16];
#pragma unroll
    for (int k = 0; k < 16; ++k) acc[k] = 0.f;
    if (h == 0) {                                   // geometric product
#pragma unroll
      for (int i = 0; i < 16; ++i)
#pragma unroll
        for (int j = 0; j < 16; ++j) {
          const int sg = (int)pga::GP.s[i][j];
          if (sg != 0) {
            float t = xr[i] * gr[j];
            acc[(int)pga::GP.k[i][j]] += (sg > 0) ? t : -t;
          }
        }
#pragma unroll
      for (int k = 0; k < 16; k += 4)
        *(float4*)(GPB + ee * LS + k) =
            make_float4(acc[k], acc[k + 1], acc[k + 2], acc[k + 3]);
    } else {                                        // equivariant join
#pragma unroll
      for (int i = 0; i < 16; ++i)
#pragma unroll
        for (int j = 0; j < 16; ++j) {
          const int sg = (int)pga::JN.s[i][j];
          if (sg != 0) {
            float t = xr[i] * gr[j];
            acc[(int)pga::JN.k[i][j]] += (sg > 0) ? t : -t;
          }
        }
      const float ps = xr[15];                      // gate by reference e0123
#pragma unroll
      for (int k = 0; k < 16; ++k) acc[k] *= ps;
#pragma unroll
      for (int k = 0; k < 16; k += 4)
        *(float4*)(JB + ee * LS + k) =
            make_float4(acc[k], acc[k + 1], acc[k + 2], acc[k + 3]);
    }
    __asm__ volatile("" ::: "memory");

    // ---- out = W2g*GP + W2j*J  (8 accumulating WMMAs) ----
    v8f O = {0.f,0.f,0.f,0.f,0.f,0.f,0.f,0.f};
    O = gemm_acc(GPB, ee, h, a2g, O);
    O = gemm_acc(JB,  ee, h, a2j, O);
    if (h == 0) O[0] += bias2;

    // transpose to element-major via LDS (reuse X slice), then coalesced store
#pragma unroll
    for (int r = 0; r < 8; r += 2) {
      v2f op2 = { O[r], O[r + 1] };
      *(v2f*)(XB + ee * LS + 8 * h + r) = op2;
    }
    __asm__ volatile("" ::: "memory");
    float4 o0 = *(const float4*)(XB + le * LS + lc);
    float4 o1 = *(const float4*)(XB + (le + 8) * LS + lc);
    float4* outv = (float4*)(out + tile * 256);
    outv[lane]      = o0;
    outv[lane + 32] = o1;
  }
}

extern "C" void kernel_launch(void* const* d_in, const int* in_sizes, int n_in,
                              void* d_out, int out_size, void* d_ws, size_t ws_size,
                              hipStream_t stream) {
  const float* x   = (const float*)d_in[0];
  const float* w1  = (const float*)d_in[1];
  const float* v1  = (const float*)d_in[2];
  const float* b1  = (const float*)d_in[3];
  const float* w2g = (const float*)d_in[4];
  const float* v2g = (const float*)d_in[5];
  const float* b2g = (const float*)d_in[6];
  const float* w2j = (const float*)d_in[7];
  const float* v2j = (const float*)d_in[8];
  const float* b2j = (const float*)d_in[9];
  float* out = (float*)d_out;

  const long long ne = (long long)in_sizes[0] / 16;   // number of multivectors
  const int tiles = (int)(ne / 16);                   // 16 MVs per wave-tile
  int blocks = (tiles + 7) / 8;                       // 8 waves (tiles) per block
  if (blocks > 1024) blocks = 1024;                   // grid-stride amortizes weights
  if (blocks < 1) blocks = 1;

  mv_ffn_kernel<<<dim3(blocks), dim3(256), 0, stream>>>(
      x, w1, v1, b1, w2g, v2g, b2g, w2j, v2j, b2j, out, tiles);
}